// SingleBranchCenterConsistencyLoss_15547781611539
// MI455X (gfx1250) — compile-verified
//
#include <hip/hip_runtime.h>
#include <hip/hip_bf16.h>

typedef __attribute__((ext_vector_type(2))) float v2f;
typedef __attribute__((ext_vector_type(8))) float v8f;

#define MOMENTUM 0.9f
#define D_DIM 256
#define DM 8

// Workspace layout (float units):
//   [0]  total accumulator
//   [1]  valid-class count accumulator
//   [64 .. 64+C*Dm)        counts per (class,domain) segment
//   [16384 .. 16384+C*Dm*D) sums -> overwritten in place with new_bank
#define WS_COUNTS_OFF 64
#define WS_BANK_OFF   16384

// ---------------------------------------------------------------- zero scratch
__global__ void k_zero(float* __restrict__ ws, int n) {
    int i = blockIdx.x * blockDim.x + threadIdx.x;
    int stride = gridDim.x * blockDim.x;
    for (; i < n; i += stride) ws[i] = 0.0f;
}

// ------------------------------------------------- segment scatter-sum (phase 1)
// One wave per row: lane loads feats[b, lane*8 .. lane*8+7] as 2x float4 and
// atomically adds into sums[seg]. Memory-bound: 64MB stream + ~67MB L2 atomics.
__global__ void k_scatter(const float* __restrict__ feats,
                          const int* __restrict__ labels,
                          const int* __restrict__ domains,
                          float* __restrict__ ws, int B) {
    int wave = threadIdx.x >> 5;
    int lane = threadIdx.x & 31;
    int b = blockIdx.x * (blockDim.x >> 5) + wave;
    if (b >= B) return;                       // wave-uniform guard

    int seg = labels[b] * DM + domains[b];
    const float4* src = (const float4*)(feats + (size_t)b * D_DIM);
    float4 v0 = src[lane * 2 + 0];
    float4 v1 = src[lane * 2 + 1];

    float* dst = ws + WS_BANK_OFF + (size_t)seg * D_DIM + lane * 8;
    atomicAdd(dst + 0, v0.x);
    atomicAdd(dst + 1, v0.y);
    atomicAdd(dst + 2, v0.z);
    atomicAdd(dst + 3, v0.w);
    atomicAdd(dst + 4, v1.x);
    atomicAdd(dst + 5, v1.y);
    atomicAdd(dst + 6, v1.z);
    atomicAdd(dst + 7, v1.w);
    if (lane == 0) atomicAdd(ws + WS_COUNTS_OFF + seg, 1.0f);
}

// --------------------------------------------------------- EMA update (phase 2)
// new_bank = present ? (mask ? 0.9*cb + 0.1*bc : bc) : cb, written in place.
__global__ void k_ema(const float* __restrict__ center_bank,
                      const unsigned char* __restrict__ center_mask,
                      float* __restrict__ ws, int CDm) {
    int i = blockIdx.x * blockDim.x + threadIdx.x;   // float4 index
    int n = CDm * (D_DIM / 4);
    if (i >= n) return;
    int s = i / (D_DIM / 4);

    float cnt = ws[WS_COUNTS_OFF + s];
    bool present = cnt > 0.0f;
    bool msk = center_mask[s] != 0;
    float inv = 1.0f / fmaxf(cnt, 1.0f);

    float4* bank = (float4*)(ws + WS_BANK_OFF);
    float4 sum = bank[i];
    float4 cb = ((const float4*)center_bank)[i];

    float4 out;
    {
        float bc, ema;
        bc = sum.x * inv; ema = MOMENTUM * cb.x + (1.0f - MOMENTUM) * bc;
        out.x = present ? (msk ? ema : bc) : cb.x;
        bc = sum.y * inv; ema = MOMENTUM * cb.y + (1.0f - MOMENTUM) * bc;
        out.y = present ? (msk ? ema : bc) : cb.y;
        bc = sum.z * inv; ema = MOMENTUM * cb.z + (1.0f - MOMENTUM) * bc;
        out.z = present ? (msk ? ema : bc) : cb.z;
        bc = sum.w * inv; ema = MOMENTUM * cb.w + (1.0f - MOMENTUM) * bc;
        out.w = present ? (msk ? ema : bc) : cb.w;
    }
    bank[i] = out;
}

// --------------------------------------------- per-class loss via WMMA (phase 3)
// One wave per class. dists[m] = diag(Y * Y^T)[m] with Y[m,d] = nb[m,d]-gc[d],
// computed with V_WMMA_F32_16X16X4_F32 accumulating over K = 256 in 64 steps.
// A-matrix layout (16x4 f32): lanes 0-15 hold M=lane with K=0 (v0) / K=1 (v1);
// lanes 16-31 hold M=lane-16 with K=2 / K=3.  B = A^T uses the same registers.
// Rows 8..15 of A are zero-padded BRANCH-FREE: every lane loads a safe row
// (m & 7) and scales by a 0/1 mask, so the inner loop has uniform control flow
// and the unrolled loads can be issued back-to-back before the WMMA chain.
__global__ void k_loss(const unsigned char* __restrict__ center_mask,
                       float* __restrict__ ws, int C) {
    __shared__ float gcbuf[8 * D_DIM];     // 8 waves per block, 256 floats each
    int wave = threadIdx.x >> 5;
    int lane = threadIdx.x & 31;
    int c = blockIdx.x * (blockDim.x >> 5) + wave;
    if (c >= C) return;                    // wave-uniform guard (EXEC stays full)

    const float* counts = ws + WS_COUNTS_OFF;
    const float* nb = ws + WS_BANK_OFF + (size_t)c * DM * D_DIM;

    // valid / present flags for the 8 domains, broadcast via ballot
    bool present_l = false, valid_l = false;
    if (lane < DM) {
        float cnt = counts[c * DM + lane];
        present_l = cnt > 0.0f;
        valid_l = present_l || (center_mask[c * DM + lane] != 0);
    }
    unsigned long long pb = __ballot(present_l);
    unsigned long long vb = __ballot(valid_l);
    unsigned validMask = (unsigned)(vb & 0xFFull);
    bool presentAny = (pb & 0xFFull) != 0;
    int nvalid = __popc(validMask);
    float invnv = 1.0f / fmaxf((float)nvalid, 1.0f);

    // global center for this class into LDS (per-wave region, same-wave DS order)
    float* gc = gcbuf + wave * D_DIM;
    #pragma unroll
    for (int j = 0; j < 8; ++j) {
        int d = lane * 8 + j;
        float s = 0.0f;
        #pragma unroll
        for (int dm = 0; dm < DM; ++dm)
            if (validMask & (1u << dm)) s += nb[dm * D_DIM + d];
        gc[d] = s * invnv;
    }

    // accumulate Y * Y^T with the matrix pipe (branch-free inner loop)
    int m = lane & 15;
    int khalf = (lane >> 4) << 1;          // 0 for lanes 0-15, 2 for lanes 16-31
    float rowmask = (m < DM) ? 1.0f : 0.0f;
    const float* rowp = nb + (size_t)(m & (DM - 1)) * D_DIM + khalf;
    const float* gcl  = gc + khalf;

    v8f acc = {};
    #pragma unroll 8
    for (int chunk = 0; chunk < D_DIM / 4; ++chunk) {
        int kb = chunk * 4;
        float b0 = rowp[kb],     b1 = rowp[kb + 1];
        float g0 = gcl[kb],      g1 = gcl[kb + 1];
        v2f a;
        a.x = (b0 - g0) * rowmask;
        a.y = (b1 - g1) * rowmask;
        acc = __builtin_amdgcn_wmma_f32_16x16x4_f32(
            false, a, false, a, (short)0, acc, false, false);
    }

    // diagonal: C[m][m] lives in acc component m at lane m (lanes 0-7 matter)
    float dval = 0.0f;
    #pragma unroll
    for (int r = 0; r < DM; ++r)
        if (lane == r) dval = acc[r];
    float contr = (lane < DM && ((validMask >> lane) & 1u)) ? dval : 0.0f;

    // wave32 reduction
    #pragma unroll
    for (int off = 16; off > 0; off >>= 1)
        contr += __shfl_xor(contr, off, 32);

    if (lane == 0) {
        bool sel = presentAny && (nvalid >= 2);
        if (sel) {
            atomicAdd(&ws[0], contr * invnv);   // class_loss
            atomicAdd(&ws[1], 1.0f);            // vc
        }
    }
}

// ------------------------------------------------------------------- finalize
__global__ void k_final(const float* __restrict__ ws, float* __restrict__ out) {
    float total = ws[0], vc = ws[1];
    out[0] = (vc > 0.0f) ? total / fmaxf(vc, 1.0f) : 0.0f;
}

extern "C" void kernel_launch(void* const* d_in, const int* in_sizes, int n_in,
                              void* d_out, int out_size, void* d_ws, size_t ws_size,
                              hipStream_t stream) {
    const float*         feats       = (const float*)d_in[0];
    const int*           labels      = (const int*)d_in[1];
    const int*           domains     = (const int*)d_in[2];
    const float*         center_bank = (const float*)d_in[3];
    const unsigned char* center_mask = (const unsigned char*)d_in[4];
    float* ws = (float*)d_ws;

    int B   = in_sizes[1];      // 65536
    int CDm = in_sizes[4];      // 8000
    int C   = CDm / DM;         // 1000

    int zeroN = WS_BANK_OFF + CDm * D_DIM;
    k_zero<<<1024, 256, 0, stream>>>(ws, zeroN);

    int rowsPerBlk = 256 / 32;  // one wave per row
    k_scatter<<<(B + rowsPerBlk - 1) / rowsPerBlk, 256, 0, stream>>>(
        feats, labels, domains, ws, B);

    int nvec = CDm * (D_DIM / 4);
    k_ema<<<(nvec + 255) / 256, 256, 0, stream>>>(center_bank, center_mask, ws, CDm);

    k_loss<<<(C + 7) / 8, 256, 0, stream>>>(center_mask, ws, C);

    k_final<<<1, 1, 0, stream>>>(ws, (float*)d_out);
}